// SimpleWithAutogradFunction_82978768158972
// MI455X (gfx1250) — compile-verified
//
#include <hip/hip_runtime.h>

// out[b] = s^2 * ( dot(x[b,:], colsum(W)) + sum(b) )   -- exact collapse of the reference
// s^2 = 0.01, x: [8192,2048] f32, W: [2048,2048] f32 ([out,in]), b: [2048] f32
// Memory-bound: ~80 MB mandatory traffic -> ~3.5us floor at 23.3 TB/s.

typedef float v2f __attribute__((ext_vector_type(2)));
typedef float v4f __attribute__((ext_vector_type(4)));
typedef float v8f __attribute__((ext_vector_type(8)));
typedef int   v4i __attribute__((ext_vector_type(4)));

#define DIM    2048
#define BATCH  8192
#define NSLICE 16
#define WSUM_OFF (NSLICE * DIM)     // float offset of wsum in ws
#define BSUM_OFF (WSUM_OFF + DIM)   // float offset of bsum in ws

#define LROWS   32                  // rows per block
#define CHUNK   128                 // K per staged chunk
#define LPAD    4
#define LSTRIDE (CHUNK + LPAD)      // 132 floats: stride%64==4 -> conflict-free gather
#define NCHUNK  (DIM / CHUNK)       // 16

// ---- CDNA5 async global->LDS path (guarded; falls back to sync staging) ----
#if __has_builtin(__builtin_amdgcn_global_load_async_to_lds_b128)
#define HAVE_ASYNC 1
#define AS1 __attribute__((address_space(1)))
#define AS3 __attribute__((address_space(3)))
__device__ __forceinline__ void cp16(const float* g, float* l) {
  // b128 async DMA: param types are <4 x i32> pointers (per clang diagnostic)
  __builtin_amdgcn_global_load_async_to_lds_b128((AS1 v4i*)g, (AS3 v4i*)l, 0, 0);
}
#if __has_builtin(__builtin_amdgcn_s_wait_asynccnt)
#define WAIT_ASYNC() __builtin_amdgcn_s_wait_asynccnt(0)
#else
#define WAIT_ASYNC() asm volatile("s_wait_asynccnt 0x0" ::: "memory")
#endif
#else
#define HAVE_ASYNC 0
__device__ __forceinline__ void cp16(const float* g, float* l) {
  *(v4f*)l = *(const v4f*)g;        // sync fallback: b128 load + ds_store_b128
}
#define WAIT_ASYNC()
#endif

// ---- Stage 1: partial column sums of W (deterministic, coalesced) ----------
__global__ __launch_bounds__(256) void colsum_partial(const float* __restrict__ W,
                                                      float* __restrict__ ws) {
  const int col = blockIdx.x * 256 + threadIdx.x;          // 0..2047
  const int o0  = blockIdx.y * (DIM / NSLICE);             // 128-row slice
  const float* p = W + (size_t)o0 * DIM + col;
  float s = 0.f;
#pragma unroll 8
  for (int o = 0; o < DIM / NSLICE; ++o) {
    __builtin_prefetch(p + (size_t)(o + 16) * DIM, 0, 1);  // global_prefetch_b8
    s += p[(size_t)o * DIM];
  }
  ws[blockIdx.y * DIM + col] = s;
}

// ---- Stage 2: finalize wsum; block 0 also reduces the bias -----------------
__global__ __launch_bounds__(256) void colsum_final(const float* __restrict__ bvec,
                                                    float* __restrict__ ws) {
  const int col = blockIdx.x * 256 + threadIdx.x;
  float s = 0.f;
#pragma unroll
  for (int sl = 0; sl < NSLICE; ++sl) s += ws[sl * DIM + col];
  ws[WSUM_OFF + col] = s;

  if (blockIdx.x == 0) {
    __shared__ float sb[256];
    float v = 0.f;
#pragma unroll
    for (int i = 0; i < DIM / 256; ++i) v += bvec[threadIdx.x + 256 * i];
    sb[threadIdx.x] = v;
    __syncthreads();
    for (int off = 128; off > 0; off >>= 1) {
      if (threadIdx.x < off) sb[threadIdx.x] += sb[threadIdx.x + off];
      __syncthreads();
    }
    if (threadIdx.x == 0) ws[BSUM_OFF] = sb[0];
  }
}

// ---- Stage 3: batched dot products on the matrix pipe ----------------------
// Block: 256 threads = 8 waves = 2 row-groups x 4 K-quarter waves; 32 rows/block.
// x tiles (32x128) double-buffer-staged into LDS by the async engine, wsum
// staged once; WMMA f32 16x16x4 with B = wsum broadcast into all 16 columns.
__global__ __launch_bounds__(256) void gemv_wmma(const float* __restrict__ x,
                                                 const float* __restrict__ ws,
                                                 float* __restrict__ out) {
  __shared__ float sx[2][LROWS * LSTRIDE];   // 33792 B
  __shared__ float sw[DIM];                  //  8192 B
  __shared__ float red[2][4][16];

  const float* wsum = ws + WSUM_OFF;
  const int tid  = threadIdx.x;
  const int wave = tid >> 5;
  const int lane = tid & 31;
  const int g    = wave >> 2;        // row group within block: 0..1
  const int q    = wave & 3;         // K quarter of each chunk
  const int m    = lane & 15;        // row within 16-row tile
  const int half = lane >> 4;        // 0: k,k+1   1: k+2,k+3 (A/B lane layout)

  const float* xblk = x + (size_t)blockIdx.x * LROWS * DIM;

  // stage wsum (8 KB) once: 512 x 16B transfers
#pragma unroll
  for (int i = 0; i < 2; ++i) {
    const int idx = i * 256 + tid;                  // float4 index 0..511
    cp16(wsum + idx * 4, &sw[idx * 4]);
  }
  // stage x chunk 0: 32 rows x 128 floats, coalesced 16B per lane
#pragma unroll
  for (int i = 0; i < 4; ++i) {
    const int idx = i * 256 + tid;                  // float4 index 0..1023
    const int row = idx >> 5, c4 = idx & 31;        // 32 float4 per row
    cp16(xblk + (size_t)row * DIM + c4 * 4, &sx[0][row * LSTRIDE + c4 * 4]);
  }
  WAIT_ASYNC();
  __syncthreads();

  v8f c = {};
  const int kq = q * (CHUNK / 4);                   // this wave's 32-K slice
  for (int ch = 0; ch < NCHUNK; ++ch) {
    const int buf = ch & 1;
    if (ch + 1 < NCHUNK) {                          // prefetch next chunk async
#pragma unroll
      for (int i = 0; i < 4; ++i) {
        const int idx = i * 256 + tid;
        const int row = idx >> 5, c4 = idx & 31;
        cp16(xblk + (size_t)row * DIM + (ch + 1) * CHUNK + c4 * 4,
             &sx[buf ^ 1][row * LSTRIDE + c4 * 4]);
      }
    }
    const float* ap = &sx[buf][(g * 16 + m) * LSTRIDE + kq + half * 2];
    const float* bp = &sw[ch * CHUNK + kq + half * 2];
#pragma unroll
    for (int kk = 0; kk < CHUNK / 4; kk += 4) {     // 8 WMMAs per chunk
      v2f a = *(const v2f*)(ap + kk);               // ds_load_b64, conflict-free
      v2f b = *(const v2f*)(bp + kk);               // ds_load_b64, broadcast
      // 8 args: (neg_a, A, neg_b, B, c_mod, C, reuse_a, reuse_b)
      c = __builtin_amdgcn_wmma_f32_16x16x4_f32(false, a, false, b,
                                                (short)0, c, false, false);
    }
    WAIT_ASYNC();        // next chunk landed (overlapped with compute above)
    __syncthreads();
  }

  // D: VGPR j = row j (lanes 0-15) / row j+8 (lanes 16-31), all N identical.
  if (lane == 0) {
#pragma unroll
    for (int j = 0; j < 8; ++j) red[g][q][j] = c[j];
  } else if (lane == 16) {
#pragma unroll
    for (int j = 0; j < 8; ++j) red[g][q][8 + j] = c[j];
  }
  __syncthreads();

  if (tid < 32) {
    const int g2 = tid >> 4, m2 = tid & 15;
    float s = red[g2][0][m2] + red[g2][1][m2] + red[g2][2][m2] + red[g2][3][m2];
    const float bs = ws[BSUM_OFF];
    out[blockIdx.x * LROWS + g2 * 16 + m2] = (s + bs) * 0.01f;  // SCALE*SCALE
  }
}

extern "C" void kernel_launch(void* const* d_in, const int* in_sizes, int n_in,
                              void* d_out, int out_size, void* d_ws, size_t ws_size,
                              hipStream_t stream) {
  const float* x = (const float*)d_in[0];   // [8192, 2048]
  const float* W = (const float*)d_in[1];   // [2048, 2048]
  const float* b = (const float*)d_in[2];   // [2048]
  float* out = (float*)d_out;               // [8192]
  float* ws  = (float*)d_ws;                // >= (16*2048 + 2048 + 1) floats ~ 136 KB

  colsum_partial<<<dim3(DIM / 256, NSLICE), 256, 0, stream>>>(W, ws);
  colsum_final<<<dim3(DIM / 256), 256, 0, stream>>>(b, ws);
  gemv_wmma<<<dim3(BATCH / LROWS), 256, 0, stream>>>(x, ws, out);
}